// BGNNDecoder_17446157156851
// MI455X (gfx1250) — compile-verified
//
#include <hip/hip_runtime.h>
#include <hip/hip_bf16.h>

typedef __attribute__((ext_vector_type(2))) float v2f;
typedef __attribute__((ext_vector_type(8))) float v8f;

// ---------------------------------------------------------------------------
// Generic fp32 WMMA GEMM:  C[M x N] = A[M x K] @ B  (B addressed via strides)
//   A row-major with leading dim lda.
//   B element [k][n] at B[k*ldbk + n*ldbn]  (ldbn=1: row-major; ldbk=1: B^T).
// One wave32 per 16x64 strip of C: 4 accumulators share one A fragment per
// k-step (4x less A traffic, 4 independent v_wmma chains per iteration).
// Requires: K % 4 == 0, N % 64 == 0 (true for all call sites here).
// Uses V_WMMA_F32_16X16X4_F32 (CDNA5 fp32 matrix op).
// ---------------------------------------------------------------------------
__global__ __launch_bounds__(32)
void gemm_wmma_f32(const float* __restrict__ A, int lda,
                   const float* __restrict__ B, int ldbk, int ldbn,
                   float* __restrict__ C, int ldc,
                   int Mdim, int Kdim) {
  const int n0   = blockIdx.x * 64;
  const int m0   = blockIdx.y * 16;
  const int lane = threadIdx.x;   // 0..31
  const int g    = lane >> 4;     // K-half select (0/1)
  const int t    = lane & 15;     // M index (A) / N index (B,C)

  int ra = m0 + t;
  if (ra >= Mdim) ra = Mdim - 1;            // clamp loads; stores are guarded
  const float* arow = A + (size_t)ra * lda;
  __builtin_prefetch(arow, 0, 1);           // global_prefetch of streaming A row

  v8f acc0 = {}, acc1 = {}, acc2 = {}, acc3 = {};
  for (int k = 0; k < Kdim; k += 4) {
    // A fragment: 16x4 f32 -> lane holds A[m][2g+v], v=0,1 (ISA 7.12.2).
    // Loaded once, reused by all 4 column tiles.
    v2f a;
    a.x = arow[k + 2 * g + 0];
    a.y = arow[k + 2 * g + 1];
    // B fragments: 4x16 f32 -> lane holds B[2g+v][n], four adjacent N-tiles.
    const float* bp = B + (size_t)(k + 2 * g) * ldbk + (size_t)(n0 + t) * ldbn;
    const size_t nstep = (size_t)16 * ldbn;
    v2f b0, b1, b2, b3;
    b0.x = bp[0];             b0.y = bp[ldbk];
    b1.x = bp[nstep];         b1.y = bp[nstep + ldbk];
    b2.x = bp[2 * nstep];     b2.y = bp[2 * nstep + ldbk];
    b3.x = bp[3 * nstep];     b3.y = bp[3 * nstep + ldbk];
    acc0 = __builtin_amdgcn_wmma_f32_16x16x4_f32(false, a, false, b0, (short)0, acc0, false, false);
    acc1 = __builtin_amdgcn_wmma_f32_16x16x4_f32(false, a, false, b1, (short)0, acc1, false, false);
    acc2 = __builtin_amdgcn_wmma_f32_16x16x4_f32(false, a, false, b2, (short)0, acc2, false, false);
    acc3 = __builtin_amdgcn_wmma_f32_16x16x4_f32(false, a, false, b3, (short)0, acc3, false, false);
  }

#pragma unroll
  for (int v = 0; v < 8; ++v) {
    int m = m0 + v + 8 * g;                 // C layout: VGPR v -> row v + 8*(lane>=16)
    if (m < Mdim) {
      float* crow = C + (size_t)m * ldc + n0 + t;
      crow[0]  = acc0[v];
      crow[16] = acc1[v];
      crow[32] = acc2[v];
      crow[48] = acc3[v];
    }
  }
}

// ---------------------------------------------------------------------------
// BatchNorm statistics: one block per column (64 columns), reduce over N rows.
// ---------------------------------------------------------------------------
__global__ void bn_stats(const float* __restrict__ x, float* __restrict__ mean,
                         float* __restrict__ rstd, int Nrows) {
  __shared__ float ssum[256];
  __shared__ float ssq[256];
  const int c = blockIdx.x;
  float s = 0.f, q = 0.f;
  for (int r = threadIdx.x; r < Nrows; r += 256) {
    float v = x[(size_t)r * 64 + c];
    s += v;
    q += v * v;
  }
  ssum[threadIdx.x] = s;
  ssq[threadIdx.x]  = q;
  __syncthreads();
  for (int off = 128; off > 0; off >>= 1) {
    if (threadIdx.x < off) {
      ssum[threadIdx.x] += ssum[threadIdx.x + off];
      ssq[threadIdx.x]  += ssq[threadIdx.x + off];
    }
    __syncthreads();
  }
  if (threadIdx.x == 0) {
    float inv = 1.0f / (float)Nrows;
    float mu  = ssum[0] * inv;
    float var = ssq[0] * inv - mu * mu;
    mean[c] = mu;
    rstd[c] = rsqrtf(var + 1e-5f);
  }
}

__global__ void bn_apply(const float* __restrict__ x,
                         const float* __restrict__ gamma,
                         const float* __restrict__ beta,
                         const float* __restrict__ mean,
                         const float* __restrict__ rstd,
                         float* __restrict__ h, int total) {
  int idx = blockIdx.x * 256 + threadIdx.x;
  if (idx >= total) return;
  int c = idx & 63;
  h[idx] = gamma[c] * (x[idx] - mean[c]) * rstd[c] + beta[c];
}

// ---------------------------------------------------------------------------
// Degree computation + inversion (rcp arrays double as accumulators).
// ---------------------------------------------------------------------------
__global__ void zero_f(float* __restrict__ p, int n) {
  int idx = blockIdx.x * 256 + threadIdx.x;
  if (idx < n) p[idx] = 0.f;
}

__global__ void deg_count(const int* __restrict__ row, const int* __restrict__ col,
                          float* __restrict__ deg_up, float* __restrict__ deg_dn, int E) {
  int e = blockIdx.x * 256 + threadIdx.x;
  if (e >= E) return;
  atomicAdd(&deg_up[col[e]], 1.0f);
  atomicAdd(&deg_dn[row[e]], 1.0f);
}

__global__ void invert_deg(float* __restrict__ p, int n) {
  int idx = blockIdx.x * 256 + threadIdx.x;
  if (idx < n) p[idx] = 1.0f / fmaxf(p[idx], 1.0f);
}

// ---------------------------------------------------------------------------
// Weight concat: Wcat[k][0:64]=Wu, [64:128]=Wd, [128:192]=Wb.
// ---------------------------------------------------------------------------
__global__ void concat_w(const float* __restrict__ Wu, const float* __restrict__ Wd,
                         const float* __restrict__ Wb, float* __restrict__ Wcat,
                         int total) {
  int idx = blockIdx.x * 256 + threadIdx.x;
  if (idx >= total) return;
  int k = idx / 192;
  int n = idx - k * 192;
  const float* src = (n < 64) ? Wu : (n < 128) ? Wd : Wb;
  Wcat[idx] = src[k * 64 + (n & 63)];
}

// cbuf[:,0:128]=0 (scatter targets), cbuf[:,128:192]=bi (copied from updnbi).
__global__ void init_concat(const float* __restrict__ updnbi,
                            float* __restrict__ cbuf, int total) {
  int idx = blockIdx.x * 256 + threadIdx.x;
  if (idx >= total) return;
  int c = idx % 192;
  cbuf[idx] = (c < 128) ? 0.f : updnbi[idx];
}

// ---------------------------------------------------------------------------
// Edge scatter: 16 threads per edge, float4 gather + scalar f32 atomics.
//   x_up[col] += up[row]   * rcp_up[col]
//   x_dn[row] += dn[col]   * rcp_dn[row]
// ---------------------------------------------------------------------------
__global__ void scatter_edges(const float* __restrict__ updnbi,
                              const int* __restrict__ row, const int* __restrict__ col,
                              const float* __restrict__ rcp_up,
                              const float* __restrict__ rcp_dn,
                              float* __restrict__ cbuf, int E) {
  long long idx = (long long)blockIdx.x * 256 + threadIdx.x;
  int e = (int)(idx >> 4);
  if (e >= E) return;
  int q = ((int)idx & 15) * 4;   // feature offset 0..60
  int r = row[e];
  int c = col[e];
  float su = rcp_up[c];
  float sd = rcp_dn[r];

  float4 u = *(const float4*)(updnbi + (size_t)r * 192 + q);
  float* du = cbuf + (size_t)c * 192 + q;
  atomicAdd(du + 0, u.x * su);
  atomicAdd(du + 1, u.y * su);
  atomicAdd(du + 2, u.z * su);
  atomicAdd(du + 3, u.w * su);

  float4 d = *(const float4*)(updnbi + (size_t)c * 192 + 64 + q);
  float* dd = cbuf + (size_t)r * 192 + 64 + q;
  atomicAdd(dd + 0, d.x * sd);
  atomicAdd(dd + 1, d.y * sd);
  atomicAdd(dd + 2, d.z * sd);
  atomicAdd(dd + 3, d.w * sd);
}

// ---------------------------------------------------------------------------
// Row L2-normalize (192-wide) + leaky ReLU, one wave32 per row.
// ---------------------------------------------------------------------------
__global__ void norm_leaky(float* __restrict__ h, int Nrows) {
  int wave = threadIdx.x >> 5;
  int lane = threadIdx.x & 31;
  int r = blockIdx.x * 8 + wave;
  if (r >= Nrows) return;
  float* p = h + (size_t)r * 192;
  float v[6];
  float s = 0.f;
#pragma unroll
  for (int j = 0; j < 6; ++j) {
    v[j] = p[lane + 32 * j];
    s += v[j] * v[j];
  }
#pragma unroll
  for (int m = 16; m > 0; m >>= 1) s += __shfl_xor(s, m, 32);
  float nrm = sqrtf(s);
  float sc  = 1.0f / fmaxf(nrm, 1e-12f);
#pragma unroll
  for (int j = 0; j < 6; ++j) {
    float t = v[j] * sc;
    p[lane + 32 * j] = (t >= 0.f) ? t : 0.1f * t;
  }
}

// ---------------------------------------------------------------------------
// Decoder gather: Ag[i]=h[di[i,0]-1], Bg[i]=h[di[i,1]-1].
// ---------------------------------------------------------------------------
__global__ void gather_ab(const float* __restrict__ h, const int* __restrict__ di,
                          float* __restrict__ Ag, float* __restrict__ Bg, int total) {
  int idx = blockIdx.x * 256 + threadIdx.x;
  if (idx >= total) return;
  int i = idx / 192;
  int f = idx - i * 192;
  int ia = di[2 * i + 0] - 1;
  int ib = di[2 * i + 1] - 1;
  Ag[idx] = h[(size_t)ia * 192 + f];
  Bg[idx] = h[(size_t)ib * 192 + f];
}

// out[i] = dot(Tg[i,:], Bg[i,:]) over 192, one wave32 per row.
__global__ void row_dot(const float* __restrict__ Tg, const float* __restrict__ Bg,
                        float* __restrict__ out, int Brows) {
  int wave = threadIdx.x >> 5;
  int lane = threadIdx.x & 31;
  int r = blockIdx.x * 8 + wave;
  if (r >= Brows) return;
  const float* tp = Tg + (size_t)r * 192;
  const float* bp = Bg + (size_t)r * 192;
  float s = 0.f;
#pragma unroll
  for (int j = 0; j < 6; ++j) s += tp[lane + 32 * j] * bp[lane + 32 * j];
#pragma unroll
  for (int m = 16; m > 0; m >>= 1) s += __shfl_xor(s, m, 32);
  if (lane == 0) out[r] = s;
}

// ---------------------------------------------------------------------------
// Host-side orchestration.
// ---------------------------------------------------------------------------
extern "C" void kernel_launch(void* const* d_in, const int* in_sizes, int n_in,
                              void* d_out, int out_size, void* d_ws, size_t ws_size,
                              hipStream_t stream) {
  const float* x     = (const float*)d_in[0];
  const int*   edge  = (const int*)d_in[1];
  const int*   drug  = (const int*)d_in[2];
  const float* gamma = (const float*)d_in[3];
  const float* beta  = (const float*)d_in[4];
  const float* Wu[3] = {(const float*)d_in[5], (const float*)d_in[8],  (const float*)d_in[11]};
  const float* Wd[3] = {(const float*)d_in[6], (const float*)d_in[9],  (const float*)d_in[12]};
  const float* Wb[3] = {(const float*)d_in[7], (const float*)d_in[10], (const float*)d_in[13]};
  const float* P1    = (const float*)d_in[14];   // 192 x 128
  const float* P2    = (const float*)d_in[15];   // 128 x 128

  const int Nn = in_sizes[0] / 64;   // 50000
  const int Ee = in_sizes[1] / 2;    // 1600000
  const int Bb = in_sizes[2] / 2;    // 4096

  const int* rowp = edge;
  const int* colp = edge + Ee;

  // Workspace carve-up (all offsets multiples of 4 floats -> 16B aligned).
  float* ws = (float*)d_ws;
  size_t o = 0;
  float* bufA   = ws + o; o += (size_t)Nn * 192;   // h (layer activations)
  float* bufB   = ws + o; o += (size_t)Nn * 192;   // updnbi (GEMM output)
  float* meanp  = ws + o; o += 64;
  float* rstd   = ws + o; o += 64;
  float* rcp_up = ws + o; o += Nn;
  float* rcp_dn = ws + o; o += Nn;
  float* Wcat   = ws + o; o += 192 * 192;
  float* T1     = ws + o; o += 192 * 128;
  float* Mm     = ws + o; o += 192 * 192;
  float* Ag     = ws + o; o += (size_t)Bb * 192;
  float* Bg     = ws + o; o += (size_t)Bb * 192;
  float* Tg     = ws + o; o += (size_t)Bb * 192;

  // Degrees (rcp arrays are contiguous: zero/invert both in one pass).
  zero_f<<<(2 * Nn + 255) / 256, 256, 0, stream>>>(rcp_up, 2 * Nn);
  deg_count<<<(Ee + 255) / 256, 256, 0, stream>>>(rowp, colp, rcp_up, rcp_dn, Ee);
  invert_deg<<<(2 * Nn + 255) / 256, 256, 0, stream>>>(rcp_up, 2 * Nn);

  // BatchNorm -> bufA (N x 64).
  bn_stats<<<64, 256, 0, stream>>>(x, meanp, rstd, Nn);
  bn_apply<<<(Nn * 64 + 255) / 256, 256, 0, stream>>>(x, gamma, beta, meanp, rstd,
                                                      bufA, Nn * 64);

  // Three BGNN conv layers.
  const int Din[3] = {64, 192, 192};
  for (int l = 0; l < 3; ++l) {
    concat_w<<<(Din[l] * 192 + 255) / 256, 256, 0, stream>>>(Wu[l], Wd[l], Wb[l],
                                                             Wcat, Din[l] * 192);
    // updnbi = h @ Wcat  (N x 192), 16x64 strip per wave.
    gemm_wmma_f32<<<dim3(192 / 64, (Nn + 15) / 16), 32, 0, stream>>>(
        bufA, Din[l], Wcat, 192, 1, bufB, 192, Nn, Din[l]);
    // cbuf: zero x_up/x_dn region, copy bi.
    init_concat<<<(Nn * 192 + 255) / 256, 256, 0, stream>>>(bufB, bufA, Nn * 192);
    // Edge scatter with f32 atomics (L2-resident on 192MB L2).
    scatter_edges<<<(int)(((long long)Ee * 16 + 255) / 256), 256, 0, stream>>>(
        bufB, rowp, colp, rcp_up, rcp_dn, bufA, Ee);
    // Row-normalize + leaky.
    norm_leaky<<<(Nn + 7) / 8, 256, 0, stream>>>(bufA, Nn);
  }

  // Decoder: M = P1 @ P2 @ P1^T.
  gemm_wmma_f32<<<dim3(128 / 64, 192 / 16), 32, 0, stream>>>(
      P1, 128, P2, 128, 1, T1, 128, 192, 128);
  gemm_wmma_f32<<<dim3(192 / 64, 192 / 16), 32, 0, stream>>>(
      T1, 128, P1, /*ldbk=*/1, /*ldbn=*/128, Mm, 192, 192, 128);  // B = P1^T

  // Gather drug embeddings, batched bilinear form.
  gather_ab<<<(Bb * 192 + 255) / 256, 256, 0, stream>>>(bufA, drug, Ag, Bg, Bb * 192);
  gemm_wmma_f32<<<dim3(192 / 64, (Bb + 15) / 16), 32, 0, stream>>>(
      Ag, 192, Mm, 192, 1, Tg, 192, Bb, 192);
  row_dot<<<(Bb + 7) / 8, 256, 0, stream>>>(Tg, Bg, (float*)d_out, Bb);
}